// GraphAttentionLayer_edit_10625749090477
// MI455X (gfx1250) — compile-verified
//
#include <hip/hip_runtime.h>
#include <stdint.h>

// Temporal graph attention (N=2048, IN=128, OUT=32) for MI455X / gfx1250.
// Bottleneck analysis: 2048^2*32 sincos pairs (~134M) dominate; att@z GEMM is
// only ~268 MFLOP -> keep full fp32 and use V_WMMA_F32_16X16X4_F32.
// z chunks are staged to LDS via the Tensor Data Mover, overlapped with the
// transcendental score phase, synced with s_wait_tensorcnt.
// Workspace requirement: ~2.5 MB (z, src, dst, per-partition max/sum/acc).

#define NN      2048
#define IN_DIM  128
#define OD      32
#define ALPHA   0.05f
#define NEG_INF -9.0e15f
#define LN_EPS  1e-6f
#define CH      256     // j-columns per partition block
#define NP      8       // partitions: NN / CH = 8

typedef float    v2f  __attribute__((ext_vector_type(2)));
typedef float    v8f  __attribute__((ext_vector_type(8)));
typedef uint32_t u32x4 __attribute__((ext_vector_type(4)));
typedef int      i32x8 __attribute__((ext_vector_type(8)));
typedef int      i32x4 __attribute__((ext_vector_type(4)));

// ---------------- Kernel 1: z = X@W, src = z@a[:32], dst = z@a[32:64] -------
__global__ __launch_bounds__(256) void k1_proj(
    const float* __restrict__ X, const float* __restrict__ W,
    const float* __restrict__ a,
    float* __restrict__ z, float* __restrict__ src, float* __restrict__ dst)
{
  int t   = threadIdx.x;
  int col = t & 31;                       // one wave32 per row
  int row = blockIdx.x * 8 + (t >> 5);
  const float* xr = X + (size_t)row * IN_DIM;
  float acc = 0.f;
#pragma unroll 4
  for (int k = 0; k < IN_DIM; ++k)
    acc = fmaf(xr[k], W[k * OD + col], acc);
  z[(size_t)row * OD + col] = acc;
  float s = acc * a[col];
  float d = acc * a[OD + col];
#pragma unroll
  for (int o = 16; o > 0; o >>= 1) { s += __shfl_xor(s, o); d += __shfl_xor(d, o); }
  if (col == 0) { src[row] = s; dst[row] = d; }
}

// ---------------- Kernel 2: per (row-tile, j-partition) attention -----------
// grid.x = (NN/16) * NP = 1024 blocks, 256 threads (8 waves).
__global__ __launch_bounds__(256) void k2_attn(
    const float* __restrict__ z,   const float* __restrict__ src,
    const float* __restrict__ dst, const int*   __restrict__ adj,
    const float* __restrict__ ts,  const int*   __restrict__ te,
    const float* __restrict__ a,   const float* __restrict__ bf,
    float* __restrict__ accP, float* __restrict__ maxP, float* __restrict__ sumP)
{
  extern __shared__ char smem[];
  float* sAtt = (float*)smem;                 // 16 x (CH+1)  (pad: stride 257 -> bank-conflict-free A loads)
  float* sZ   = sAtt + 16 * (CH + 1);         // CH x 32 (TDM destination)
  float* sSrc = sZ + CH * OD;                 // 16
  float* sMax = sSrc + 16;                    // 16

  const int t    = threadIdx.x;
  const int wave = t >> 5;
  const int lane = t & 31;
  const int tile = blockIdx.x >> 3;           // row tile (16 rows)
  const int p    = blockIdx.x & (NP - 1);     // j partition
  const int rowbase = tile * 16;
  const int jbase   = p * CH;
  const float tef = (float)te[0];

  // --- Tensor Data Mover: async copy z[jbase .. jbase+CH) x 32 -> sZ -------
  if (wave == 0) {
    uint64_t ga = (uint64_t)(uintptr_t)(z + (size_t)jbase * OD);
    uint32_t lz = (uint32_t)(uintptr_t)sZ;    // low 32 bits of LDS flat addr = LDS byte offset
    u32x4 g0;
    g0[0] = 1u;                                         // count=1, user descriptor
    g0[1] = lz;                                         // lds_addr
    g0[2] = (uint32_t)ga;                               // global_addr[31:0]
    g0[3] = (uint32_t)((ga >> 32) & 0x1FFFFFFull) | (2u << 30); // addr[56:32] | type=2
    i32x8 g1;
    g1[0] = (2 << 16);                 // data_size = 4 bytes
    g1[1] = (int)(32u << 16);          // tensor_dim0 = 32 (bits 79:48, low half)
    g1[2] = (int)((uint32_t)CH << 16); // tensor_dim1 = CH (bits 111:80, low half)
    g1[3] = (int)(32u << 16);          // tile_dim0 = 32 (bits 127:112)
    g1[4] = CH;                        // tile_dim1 = CH, tile_dim2 = 0 (2D)
    g1[5] = 32;                        // tensor_dim0_stride = 32
    g1[6] = 0;                         // tensor_dim1_stride unused for 2D
    g1[7] = 0;
    i32x4 g2 = {0, 0, 0, 0};
    i32x4 g3 = {0, 0, 0, 0};
    i32x8 g4 = {0, 0, 0, 0, 0, 0, 0, 0};
    __builtin_amdgcn_tensor_load_to_lds(g0, g1, g2, g3, g4, 0);
  }
  if (t < 16) sSrc[t] = src[rowbase + t];

  // --- score phase: thread t owns column jj = t of all 16 rows ------------
  float dtv[16], et[16];
#pragma unroll
  for (int r = 0; r < 16; ++r) {
    dtv[r] = tef - ts[(size_t)(rowbase + r) * NN + jbase + t];
    et[r]  = 0.f;
  }
  for (int k = 0; k < OD; ++k) {      // 134M sincos total across grid: the hot loop
    float f  = bf[k];
    float as = a[2 * k];
    float ac = a[2 * k + 1];
#pragma unroll
    for (int r = 0; r < 16; ++r) {
      float sn, cs;
      __sincosf(dtv[r] * f, &sn, &cs);
      et[r] = fmaf(sn, as, fmaf(cs, ac, et[r]));
    }
  }
  if (wave == 0) __builtin_amdgcn_s_wait_tensorcnt(0);  // TDM overlapped with sincos
  __syncthreads();                                      // sSrc + sZ visible

  float dstv = dst[jbase + t];
#pragma unroll
  for (int r = 0; r < 16; ++r) {
    int   ad = adj[(size_t)(rowbase + r) * NN + jbase + t];
    float e  = sSrc[r] + dstv + et[r];
    e = (e > 0.f) ? e : ALPHA * e;                      // leaky relu
    sAtt[r * (CH + 1) + t] = (ad > 0) ? e : NEG_INF;
  }
  __syncthreads();

  // --- per-row max (wave w handles rows w, w+8) ---------------------------
  for (int rr = wave; rr < 16; rr += 8) {
    float m = -3.0e38f;
    for (int c = lane; c < CH; c += 32) m = fmaxf(m, sAtt[rr * (CH + 1) + c]);
#pragma unroll
    for (int o = 16; o > 0; o >>= 1) m = fmaxf(m, __shfl_xor(m, o));
    if (lane == 0) { sMax[rr] = m; maxP[(size_t)(rowbase + rr) * NP + p] = m; }
  }
  __syncthreads();

  // --- exp in place -------------------------------------------------------
#pragma unroll
  for (int r = 0; r < 16; ++r) {
    float pv = __expf(sAtt[r * (CH + 1) + t] - sMax[r]);
    sAtt[r * (CH + 1) + t] = pv;
  }
  __syncthreads();

  // --- per-row sum --------------------------------------------------------
  for (int rr = wave; rr < 16; rr += 8) {
    float sm = 0.f;
    for (int c = lane; c < CH; c += 32) sm += sAtt[rr * (CH + 1) + c];
#pragma unroll
    for (int o = 16; o > 0; o >>= 1) sm += __shfl_xor(sm, o);
    if (lane == 0) sumP[(size_t)(rowbase + rr) * NP + p] = sm;
  }

  // --- GEMM: P[16xCH] @ Zc[CHx32] via V_WMMA_F32_16X16X4_F32 --------------
  // waves 0/1 each own one 16-wide N tile; K = CH in steps of 4.
  if (wave < 2) {
    const int nt  = wave;
    const int m16 = lane & 15;
    const int hi  = lane >> 4;
    v8f acc = {0.f, 0.f, 0.f, 0.f, 0.f, 0.f, 0.f, 0.f};
    for (int kk = 0; kk < CH; kk += 4) {
      int ka = kk + 2 * hi;                       // A: lane holds M=m16, K = 2*hi + v
      v2f av, bv;
      av.x = sAtt[m16 * (CH + 1) + ka];
      av.y = sAtt[m16 * (CH + 1) + ka + 1];
      bv.x = sZ[(size_t)ka * OD + nt * 16 + m16]; // B: lane holds N=m16, K = 2*hi + v
      bv.y = sZ[(size_t)(ka + 1) * OD + nt * 16 + m16];
      acc = __builtin_amdgcn_wmma_f32_16x16x4_f32(
          /*neg_a=*/false, av, /*neg_b=*/false, bv,
          /*c_mod=*/(short)0, acc, /*reuse_a=*/false, /*reuse_b=*/false);
    }
    // C/D layout: vgpr v, lane l -> row = v + 8*(l/16), col = l%16
#pragma unroll
    for (int v = 0; v < 8; ++v) {
      int row = v + 8 * hi;
      int col = nt * 16 + m16;
      accP[((size_t)(rowbase + row) * NP + p) * OD + col] = acc[v];
    }
  }
}

// ---------------- Kernel 3: merge partitions, residual, LayerNorm ----------
__global__ __launch_bounds__(256) void k3_combine(
    const float* __restrict__ z,    const float* __restrict__ accP,
    const float* __restrict__ maxP, const float* __restrict__ sumP,
    const float* __restrict__ gam,  const float* __restrict__ bet,
    float* __restrict__ out)
{
  int t   = threadIdx.x;
  int col = t & 31;                      // one wave32 per row
  int row = blockIdx.x * 8 + (t >> 5);
  float mp[NP];
  float M = -3.0e38f;
#pragma unroll
  for (int p = 0; p < NP; ++p) { mp[p] = maxP[(size_t)row * NP + p]; M = fmaxf(M, mp[p]); }
  float S = 0.f, x = 0.f;
#pragma unroll
  for (int p = 0; p < NP; ++p) {
    float w = __expf(mp[p] - M);
    S += sumP[(size_t)row * NP + p] * w;
    x += accP[((size_t)row * NP + p) * OD + col] * w;
  }
  x = x / S + z[(size_t)row * OD + col];        // softmax normalize + residual
  float s1 = x;
#pragma unroll
  for (int o = 16; o > 0; o >>= 1) s1 += __shfl_xor(s1, o);
  float mu = s1 * (1.0f / OD);
  float d  = x - mu;
  float s2 = d * d;
#pragma unroll
  for (int o = 16; o > 0; o >>= 1) s2 += __shfl_xor(s2, o);
  float var = s2 * (1.0f / OD);
  out[(size_t)row * OD + col] = d * rsqrtf(var + LN_EPS) * gam[col] + bet[col];
}

// ---------------- host launcher --------------------------------------------
extern "C" void kernel_launch(void* const* d_in, const int* in_sizes, int n_in,
                              void* d_out, int out_size, void* d_ws, size_t ws_size,
                              hipStream_t stream) {
  const float* X   = (const float*)d_in[0];   // ini_feature [2048,128]
  const int*   adj = (const int*)  d_in[1];   // adj [2048,2048]
  const float* ts  = (const float*)d_in[2];   // timestamp [2048,2048]
  const float* a   = (const float*)d_in[3];   // a [64]
  const int*   te  = (const int*)  d_in[4];   // time_end scalar
  const float* W   = (const float*)d_in[5];   // W [128,32]
  const float* bf  = (const float*)d_in[6];   // basis_freq [32]
  const float* gam = (const float*)d_in[7];   // ln_gamma [32]
  const float* bet = (const float*)d_in[8];   // ln_beta [32]
  float* out = (float*)d_out;

  float* ws   = (float*)d_ws;                 // needs ~2.5 MB
  float* z    = ws;                           // NN*OD      = 65536
  float* src  = z    + (size_t)NN * OD;       // NN
  float* dst  = src  + NN;                    // NN
  float* maxP = dst  + NN;                    // NN*NP
  float* sumP = maxP + (size_t)NN * NP;       // NN*NP
  float* accP = sumP + (size_t)NN * NP;       // NN*NP*OD

  k1_proj<<<NN / 8, 256, 0, stream>>>(X, W, a, z, src, dst);

  size_t sh2 = (size_t)(16 * (CH + 1) + CH * OD + 16 + 16) * sizeof(float); // ~48 KB
  k2_attn<<<(NN / 16) * NP, 256, sh2, stream>>>(z, src, dst, adj, ts, te, a, bf,
                                                accP, maxP, sumP);

  k3_combine<<<NN / 8, 256, 0, stream>>>(z, accP, maxP, sumP, gam, bet, out);
}